// StaticGAT_43267500540700
// MI455X (gfx1250) — compile-verified
//
#include <hip/hip_runtime.h>
#include <math.h>

typedef __attribute__((ext_vector_type(2))) float v2f;
typedef __attribute__((ext_vector_type(8))) float v8f;

#define N_NODES 50000
#define F_DIM   128
#define E_RAW   800000
#define E_TOT   (E_RAW + N_NODES)
#define NEG_SLOPE 0.2f

// ---------------------------------------------------------------------------
// GEMM: Y[N,128] = X[N,128] @ W[128,128] + b   via V_WMMA_F32_16X16X4_F32
// block = 128 threads (4 waves). grid = (N/16, 2). wave w covers N-tile
// (blockIdx.y*4 + w). 32 k-steps of 16x16x4.
// ---------------------------------------------------------------------------
__global__ __launch_bounds__(128)
void gat_gemm_wmma(const float* __restrict__ X, const float* __restrict__ W,
                   const float* __restrict__ bias, float* __restrict__ Y, int N)
{
    const int wave = threadIdx.x >> 5;
    const int lane = threadIdx.x & 31;
    const int lo   = lane & 15;
    const int hi   = lane >> 4;                 // 0 or 1
    const int m0   = blockIdx.x * 16;
    const int n0   = (blockIdx.y * 4 + wave) * 16;
    if (m0 >= N) return;

    v8f acc = {};
    const float* __restrict__ xrow = X + (size_t)(m0 + lo) * F_DIM;
#pragma unroll 8
    for (int k0 = 0; k0 < F_DIM; k0 += 4) {
        // A 16x4: lane<16 -> K = k0,k0+1 ; lane>=16 -> K = k0+2,k0+3 ; M = lo
        v2f a, b;
        const int ka = k0 + 2 * hi;
        a.x = xrow[ka];
        a.y = xrow[ka + 1];
        // B 4x16: N = n0+lo ; K = k0 + 2*hi + vgpr
        b.x = W[(size_t)(k0 + 2 * hi) * F_DIM + n0 + lo];
        b.y = W[(size_t)(k0 + 2 * hi + 1) * F_DIM + n0 + lo];
        acc = __builtin_amdgcn_wmma_f32_16x16x4_f32(
            false, a, false, b, (short)0, acc, false, false);
    }
    const float bv = bias[n0 + lo];
#pragma unroll
    for (int r = 0; r < 8; ++r) {
        // D: VGPR r -> M = m0 + r + 8*hi, col = n0 + lo
        Y[(size_t)(m0 + r + 8 * hi) * F_DIM + n0 + lo] = acc[r] + bv;
    }
}

// ---------------------------------------------------------------------------
// Per-layer scratch init: m = -inf, denom = 0, acc = 0
// ---------------------------------------------------------------------------
__global__ __launch_bounds__(256)
void gat_init(float* __restrict__ m, float* __restrict__ denom,
              float* __restrict__ acc, int NH, int NF)
{
    const int i = blockIdx.x * 256 + threadIdx.x;
    if (i < NH) { m[i] = -__builtin_inff(); denom[i] = 0.0f; }
    if (i < NF) acc[i] = 0.0f;
}

__device__ __forceinline__ void atomicMaxF32(float* addr, float v)
{
    // bit-pattern trick: valid for any mix given -inf init
    if (v >= 0.0f) atomicMax((int*)addr, __float_as_int(v));
    else           atomicMin((unsigned int*)addr, __float_as_uint(v));
}

// ---------------------------------------------------------------------------
// Edge scores + segment max.  One wave per edge; lane handles 4 feature dims.
// score[e,h] = att[h,:] . leaky_relu(xl[src] + xr[dst])[h,:]
// ---------------------------------------------------------------------------
__global__ __launch_bounds__(256)
void gat_edge_score(const float* __restrict__ xl, const float* __restrict__ xr,
                    const long long* __restrict__ esrc,
                    const long long* __restrict__ edst,
                    const float* __restrict__ att,
                    float* __restrict__ score, float* __restrict__ mbuf,
                    int E0, int Etot, int H)
{
    const int e = blockIdx.x * 8 + (threadIdx.x >> 5);
    if (e >= Etot) return;
    const int lane = threadIdx.x & 31;

    int src, dst;
    if (e < E0) { src = (int)esrc[e]; dst = (int)edst[e]; }
    else        { src = dst = e - E0; }

    const float4 a = ((const float4*)(xl + (size_t)src * F_DIM))[lane];
    const float4 b = ((const float4*)(xr + (size_t)dst * F_DIM))[lane];
    const float4 w = ((const float4*)att)[lane];

    float sx = a.x + b.x; sx = sx > 0.0f ? sx : NEG_SLOPE * sx;
    float sy = a.y + b.y; sy = sy > 0.0f ? sy : NEG_SLOPE * sy;
    float sz = a.z + b.z; sz = sz > 0.0f ? sz : NEG_SLOPE * sz;
    float sw = a.w + b.w; sw = sw > 0.0f ? sw : NEG_SLOPE * sw;
    float part = sx * w.x + sy * w.y + sz * w.z + sw * w.w;

    const int gs = 32 / H;                       // lanes per head (8 or 32)
    for (int off = 1; off < gs; off <<= 1)
        part += __shfl_xor(part, off, 32);

    if ((lane & (gs - 1)) == 0) {
        const int h = lane / gs;
        score[(size_t)e * H + h] = part;
        atomicMaxF32(&mbuf[(size_t)dst * H + h], part);
    }
}

// ---------------------------------------------------------------------------
// Edge exp + scatter: ex = exp(score - m[dst]); denom[dst,h] += ex;
// acc[dst,:] += ex[head(d)] * xl[src,:]
// ---------------------------------------------------------------------------
__global__ __launch_bounds__(256)
void gat_edge_scatter(const float* __restrict__ xl,
                      const float* __restrict__ score,
                      const float* __restrict__ mbuf,
                      const long long* __restrict__ esrc,
                      const long long* __restrict__ edst,
                      float* __restrict__ denom, float* __restrict__ acc,
                      int E0, int Etot, int H)
{
    const int e = blockIdx.x * 8 + (threadIdx.x >> 5);
    if (e >= Etot) return;
    const int lane = threadIdx.x & 31;

    int src, dst;
    if (e < E0) { src = (int)esrc[e]; dst = (int)edst[e]; }
    else        { src = dst = e - E0; }

    const int gs = 32 / H;
    const int h  = lane / gs;
    const float ex = __expf(score[(size_t)e * H + h] - mbuf[(size_t)dst * H + h]);

    if ((lane & (gs - 1)) == 0)
        atomicAdd(&denom[(size_t)dst * H + h], ex);

    const float4 v = ((const float4*)(xl + (size_t)src * F_DIM))[lane];
    float* pa = acc + (size_t)dst * F_DIM + lane * 4;
    atomicAdd(pa + 0, ex * v.x);
    atomicAdd(pa + 1, ex * v.y);
    atomicAdd(pa + 2, ex * v.z);
    atomicAdd(pa + 3, ex * v.w);
}

// ---------------------------------------------------------------------------
// Per-node normalize + bias (+ ELU between layers)
// ---------------------------------------------------------------------------
__global__ __launch_bounds__(256)
void gat_finalize(const float* __restrict__ acc, const float* __restrict__ denom,
                  const float* __restrict__ bias, float* __restrict__ out,
                  int N, int H, int do_elu)
{
    const int i = blockIdx.x * 256 + threadIdx.x;
    if (i >= N * F_DIM) return;
    const int node = i >> 7;
    const int d    = i & (F_DIM - 1);
    const int h    = d / (F_DIM / H);
    float v = acc[i] / denom[(size_t)node * H + h] + bias[d];
    if (do_elu) v = v > 0.0f ? v : (__expf(v) - 1.0f);
    out[i] = v;
}

// ---------------------------------------------------------------------------
extern "C" void kernel_launch(void* const* d_in, const int* in_sizes, int n_in,
                              void* d_out, int out_size, void* d_ws, size_t ws_size,
                              hipStream_t stream)
{
    (void)out_size; (void)ws_size;
    const int N = N_NODES, F = F_DIM, E0 = E_RAW, Etot = E_TOT;

    // Locate inputs by size (robust to pytree flattening order).
    int xi = -1, ei = -1, pidx[18], np = 0;
    for (int i = 0; i < n_in; ++i) {
        if (in_sizes[i] == N * F && xi < 0)      xi = i;
        else if (in_sizes[i] == 2 * E0 && ei < 0) ei = i;
        else if (np < 18)                         pidx[np++] = i;
    }
    const float*     x    = (const float*)d_in[xi];
    const long long* eidx = (const long long*)d_in[ei];
    const long long* esrc = eidx;
    const long long* edst = eidx + E0;

    // Workspace layout (floats)
    float* ws = (float*)d_ws;
    size_t o = 0;
    float* hbuf  = ws + o; o += (size_t)N * F;
    float* xl    = ws + o; o += (size_t)N * F;
    float* xr    = ws + o; o += (size_t)N * F;
    float* score = ws + o; o += (size_t)Etot * 4;
    float* mbuf  = ws + o; o += (size_t)N * 4;
    float* denom = ws + o; o += (size_t)N * 4;
    float* acc   = ws + o; o += (size_t)N * F;

    const int Hs[3] = {4, 4, 1};
    const float* in = x;

    for (int l = 0; l < 3; ++l) {
        const int base = 6 * l;
        const float *Wl, *Wr, *att, *bl, *br, *bias;
        // Detect param ordering per layer: insertion (Wl,bl,Wr,br,att,bias)
        // vs alphabetical-pytree (Wl,Wr,att,bias,bl,br). Wl/Wr are 16384 elems.
        if (in_sizes[pidx[base + 1]] == F * F) {
            Wl   = (const float*)d_in[pidx[base + 0]];
            Wr   = (const float*)d_in[pidx[base + 1]];
            att  = (const float*)d_in[pidx[base + 2]];
            bias = (const float*)d_in[pidx[base + 3]];
            bl   = (const float*)d_in[pidx[base + 4]];
            br   = (const float*)d_in[pidx[base + 5]];
        } else {
            Wl   = (const float*)d_in[pidx[base + 0]];
            bl   = (const float*)d_in[pidx[base + 1]];
            Wr   = (const float*)d_in[pidx[base + 2]];
            br   = (const float*)d_in[pidx[base + 3]];
            att  = (const float*)d_in[pidx[base + 4]];
            bias = (const float*)d_in[pidx[base + 5]];
        }
        const int H = Hs[l];
        float* out = (l == 2) ? (float*)d_out : hbuf;   // in-place hbuf reuse
                                                        // is stream-order safe:
                                                        // GEMMs (only readers of
                                                        // `in`) precede finalize.
        gat_init<<<(N * F + 255) / 256, 256, 0, stream>>>(mbuf, denom, acc,
                                                          N * H, N * F);
        dim3 ggrid(N / 16, 2);
        gat_gemm_wmma<<<ggrid, 128, 0, stream>>>(in, Wl, bl, xl, N);
        gat_gemm_wmma<<<ggrid, 128, 0, stream>>>(in, Wr, br, xr, N);

        const int eb = (Etot + 7) / 8;
        gat_edge_score<<<eb, 256, 0, stream>>>(xl, xr, esrc, edst, att,
                                               score, mbuf, E0, Etot, H);
        gat_edge_scatter<<<eb, 256, 0, stream>>>(xl, score, mbuf, esrc, edst,
                                                 denom, acc, E0, Etot, H);
        gat_finalize<<<(N * F + 255) / 256, 256, 0, stream>>>(acc, denom, bias,
                                                              out, N, H, l < 2);
        in = hbuf;
    }
}